// HPWLAtomic_33767032881790
// MI455X (gfx1250) — compile-verified
//
#include <hip/hip_runtime.h>

// ---------------------------------------------------------------------------
// HPWL: segment min/max over pins -> per-net half-perimeter -> masked sum.
// MI455X strategy:
//   - ordered-uint min/max atomics, resident in the 192MB L2 (48MB state)
//   - stream-once inputs (pos, pin2net, mask) loaded with TH=NT so they do
//     not evict the atomic working set from L2
//   - 128-bit loads everywhere on the hot path; wave32 shuffle reductions
//   ws layout: [xmin_u | xmax_u | ymin_u | ymax_u], each num_nets u32.
// ---------------------------------------------------------------------------

// clang vector types: required by __builtin_nontemporal_load
typedef float         v4f  __attribute__((ext_vector_type(4)));
typedef int           v4i  __attribute__((ext_vector_type(4)));
typedef unsigned int  v4u  __attribute__((ext_vector_type(4)));
typedef unsigned char v4b  __attribute__((ext_vector_type(4)));

// Monotonic float <-> uint mapping: ordf(a) < ordf(b)  <=>  a < b
__device__ __forceinline__ unsigned ordf(float f) {
    unsigned u = __float_as_uint(f);
    return (u & 0x80000000u) ? ~u : (u | 0x80000000u);
}
__device__ __forceinline__ float unordf(unsigned u) {
    return __uint_as_float((u & 0x80000000u) ? (u & 0x7fffffffu) : ~u);
}

__global__ void hpwl_init(unsigned* __restrict__ ws, int num_nets,
                          float* __restrict__ out) {
    int i = blockIdx.x * blockDim.x + threadIdx.x;
    if (i == 0) out[0] = 0.0f;              // d_out is poisoned; re-zero every call
    unsigned* xmin_u = ws;
    unsigned* xmax_u = ws + (size_t)num_nets;
    unsigned* ymin_u = ws + 2 * (size_t)num_nets;
    unsigned* ymax_u = ws + 3 * (size_t)num_nets;
    int stride = gridDim.x * blockDim.x;
    for (int n = i; n < num_nets; n += stride) {
        xmin_u[n] = 0xFFFFFFFFu;   // identity for ordered-min
        xmax_u[n] = 0u;            // identity for ordered-max; also "untouched" marker
        ymin_u[n] = 0xFFFFFFFFu;
        ymax_u[n] = 0u;
    }
}

__device__ __forceinline__ void scatter_one(int net, float px, float py,
                                            unsigned* __restrict__ xmin_u,
                                            unsigned* __restrict__ xmax_u,
                                            unsigned* __restrict__ ymin_u,
                                            unsigned* __restrict__ ymax_u) {
    unsigned ox = ordf(px), oy = ordf(py);
    atomicMin(&xmin_u[net], ox);   // global_atomic_min_u32 (non-returning, L2-resident)
    atomicMax(&xmax_u[net], ox);   // global_atomic_max_u32
    atomicMin(&ymin_u[net], oy);
    atomicMax(&ymax_u[net], oy);
}

// One thread handles 4 consecutive pins via 128-bit non-temporal loads.
__global__ void hpwl_scatter(const float* __restrict__ x,
                             const float* __restrict__ y,
                             const int* __restrict__ p2n,
                             unsigned* __restrict__ ws,
                             int num_pins, int num_nets) {
    unsigned* xmin_u = ws;
    unsigned* xmax_u = ws + (size_t)num_nets;
    unsigned* ymin_u = ws + 2 * (size_t)num_nets;
    unsigned* ymax_u = ws + 3 * (size_t)num_nets;

    int t    = blockIdx.x * blockDim.x + threadIdx.x;
    int base = t * 4;
    bool vec_ok = ((num_pins & 3) == 0);   // y = pos+num_pins stays 16B aligned

    if (vec_ok && base + 3 < num_pins) {
        // stream-once data: TH=NT so it does not evict the atomic arrays from L2
        v4f xv = __builtin_nontemporal_load(reinterpret_cast<const v4f*>(x + base));
        v4f yv = __builtin_nontemporal_load(reinterpret_cast<const v4f*>(y + base));
        v4i nv = __builtin_nontemporal_load(reinterpret_cast<const v4i*>(p2n + base));
        scatter_one(nv.x, xv.x, yv.x, xmin_u, xmax_u, ymin_u, ymax_u);
        scatter_one(nv.y, xv.y, yv.y, xmin_u, xmax_u, ymin_u, ymax_u);
        scatter_one(nv.z, xv.z, yv.z, xmin_u, xmax_u, ymin_u, ymax_u);
        scatter_one(nv.w, xv.w, yv.w, xmin_u, xmax_u, ymin_u, ymax_u);
    } else {
        for (int p = base; p < num_pins && p < base + 4; ++p) {
            scatter_one(p2n[p], x[p], y[p], xmin_u, xmax_u, ymin_u, ymax_u);
        }
    }
}

// One thread handles 4 consecutive nets via 128-bit loads, then wave32
// shuffle-tree reduction + one f32 atomic add per block.
__global__ void hpwl_reduce(const unsigned* __restrict__ ws,
                            const unsigned char* __restrict__ mask,
                            int num_nets, float* __restrict__ out) {
    const unsigned* xmin_u = ws;
    const unsigned* xmax_u = ws + (size_t)num_nets;
    const unsigned* ymin_u = ws + 2 * (size_t)num_nets;
    const unsigned* ymax_u = ws + 3 * (size_t)num_nets;

    int t    = blockIdx.x * blockDim.x + threadIdx.x;
    int base = t * 4;
    bool vec_ok = ((num_nets & 3) == 0);

    float local = 0.0f;
    if (vec_ok && base + 3 < num_nets) {
        // ws arrays: expected L2 hits from the scatter pass -> regular temporal
        v4u xM = *reinterpret_cast<const v4u*>(xmax_u + base);
        v4u xm = *reinterpret_cast<const v4u*>(xmin_u + base);
        v4u yM = *reinterpret_cast<const v4u*>(ymax_u + base);
        v4u ym = *reinterpret_cast<const v4u*>(ymin_u + base);
        // mask: stream-once -> NT
        v4b mk = __builtin_nontemporal_load(reinterpret_cast<const v4b*>(mask + base));
        if (mk.x && xM.x) local += (unordf(xM.x) - unordf(xm.x)) + (unordf(yM.x) - unordf(ym.x));
        if (mk.y && xM.y) local += (unordf(xM.y) - unordf(xm.y)) + (unordf(yM.y) - unordf(ym.y));
        if (mk.z && xM.z) local += (unordf(xM.z) - unordf(xm.z)) + (unordf(yM.z) - unordf(ym.z));
        if (mk.w && xM.w) local += (unordf(xM.w) - unordf(xm.w)) + (unordf(yM.w) - unordf(ym.w));
    } else {
        for (int n = base; n < num_nets && n < base + 4; ++n) {
            unsigned xM = xmax_u[n];
            if (mask[n] && xM != 0u) {
                local += (unordf(xM) - unordf(xmin_u[n])) +
                         (unordf(ymax_u[n]) - unordf(ymin_u[n]));
            }
        }
    }

    // wave32 butterfly reduction
    #pragma unroll
    for (int off = 16; off > 0; off >>= 1)
        local += __shfl_xor(local, off, 32);

    __shared__ float smem[8];                 // 256 threads = 8 wave32
    int lane = threadIdx.x & 31;
    int wave = threadIdx.x >> 5;
    if (lane == 0) smem[wave] = local;
    __syncthreads();
    if (wave == 0) {
        float v = (lane < (blockDim.x >> 5)) ? smem[lane] : 0.0f;
        #pragma unroll
        for (int off = 4; off > 0; off >>= 1)
            v += __shfl_xor(v, off, 32);
        if (lane == 0)
            __hip_atomic_fetch_add(out, v, __ATOMIC_RELAXED,
                                   __HIP_MEMORY_SCOPE_AGENT);  // global_atomic_add_f32
    }
}

extern "C" void kernel_launch(void* const* d_in, const int* in_sizes, int n_in,
                              void* d_out, int out_size, void* d_ws, size_t ws_size,
                              hipStream_t stream) {
    const float*         pos  = (const float*)d_in[0];
    const int*           p2n  = (const int*)d_in[1];
    const unsigned char* mask = (const unsigned char*)d_in[2];
    const int num_pins = in_sizes[1];   // pin2net_map length
    const int num_nets = in_sizes[2];   // net_mask length

    const float* x = pos;
    const float* y = pos + num_pins;
    unsigned* ws  = (unsigned*)d_ws;    // needs 4 * num_nets * 4 bytes (48 MB)
    float*    out = (float*)d_out;

    const int BS = 256;

    int init_blocks = (num_nets + BS - 1) / BS;
    hpwl_init<<<init_blocks, BS, 0, stream>>>(ws, num_nets, out);

    int sthreads = (num_pins + 3) / 4;
    int sblocks  = (sthreads + BS - 1) / BS;
    hpwl_scatter<<<sblocks, BS, 0, stream>>>(x, y, p2n, ws, num_pins, num_nets);

    int rthreads = (num_nets + 3) / 4;
    int rblocks  = (rthreads + BS - 1) / BS;
    hpwl_reduce<<<rblocks, BS, 0, stream>>>(ws, mask, num_nets, out);
}